// Block_57844619543215
// MI455X (gfx1250) — compile-verified
//
#include <hip/hip_runtime.h>
#include <math.h>

#define NN   30000
#define EE   300000
#define DIM  64
#define EDIM 16
#define HH   3
#define TT   6

#define KK     64          // K of every GEMM in this model
#define BSTR   (KK + 4)    // LDS row stride (pad: bank = (4*col + k) % 64, conflict-free)
#define MWAVES 5           // 1875 M-tiles = 5 * 375  -> no tail, EXEC all-ones

typedef __attribute__((ext_vector_type(2))) float v2f;
typedef __attribute__((ext_vector_type(8))) float v8f;

// ---------------- utility ----------------
__global__ void k_fill(float* p, float v, int n) {
    int i = blockIdx.x * blockDim.x + threadIdx.x;
    if (i < n) p[i] = v;
}

__device__ inline void atomicMaxF(float* addr, float val) {
    unsigned int* ua = (unsigned int*)addr;
    unsigned int old = *ua;
    while (__uint_as_float(old) < val) {
        unsigned int assumed = old;
        old = atomicCAS(ua, assumed, __float_as_uint(val));
        if (old == assumed) break;
    }
}

// ---------------- WMMA fp32 GEMM (K=64) ----------------
// C[M x ldc] strip of 64 columns starting at n0 = blockIdx.y*64.
// Block = 5 waves; wave w owns M-tile m0 = (blockIdx.x*5 + w)*16 and computes a
// 16x64 strip with 4 accumulators. B strip (64 cols x 64 k) staged in LDS,
// [col][k] layout (stride BSTR), so B fragments are contiguous ds_load_b64.
// fp32 WMMA frag mapping (ISA 7.12.2), r = lane&15, g = lane>>4:
//   A: {A[m0+r][k+2g], A[m0+r][k+2g+1]}   B: {B[k+2g][col], B[k+2g+1][col]}
//   C: acc[v] = C[m0+v+8g][col]
template <bool TRANSB, bool ACCUM, bool HASBIAS>
__global__ __launch_bounds__(MWAVES * 32)
void k_wmma_gemm(const float* __restrict__ A, const float* __restrict__ B,
                 const float* __restrict__ bias, float* __restrict__ C,
                 int ldb, int ldc)
{
    __shared__ float Bs[64 * BSTR];
    const int tid  = threadIdx.x;
    const int wave = tid >> 5;
    const int lane = tid & 31;
    const int g = lane >> 4;
    const int r = lane & 15;
    const int n0 = blockIdx.y * 64;
    const int m0 = (blockIdx.x * MWAVES + wave) * 16;

    // cooperative load of B strip into LDS as Bs[col][k]
    for (int i = tid; i < 64 * KK; i += MWAVES * 32) {
        int col, k;
        if (TRANSB) { col = i >> 6; k = i & (KK - 1); }   // consecutive i -> consecutive k
        else        { col = i & 63; k = i >> 6;       }   // consecutive i -> consecutive col
        Bs[col * BSTR + k] = TRANSB ? B[(size_t)(n0 + col) * ldb + k]
                                    : B[(size_t)k * ldb + (n0 + col)];
    }
    __syncthreads();

    v8f acc0 = {}, acc1 = {}, acc2 = {}, acc3 = {};
    const float* Arow = A + (size_t)(m0 + r) * KK;
#pragma unroll
    for (int k = 0; k < KK; k += 4) {
        const int ka = k + 2 * g;                 // even -> 8B aligned v2f loads
        v2f a  = *(const v2f*)&Arow[ka];
        v2f b0 = *(const v2f*)&Bs[(r +  0) * BSTR + ka];
        v2f b1 = *(const v2f*)&Bs[(r + 16) * BSTR + ka];
        v2f b2 = *(const v2f*)&Bs[(r + 32) * BSTR + ka];
        v2f b3 = *(const v2f*)&Bs[(r + 48) * BSTR + ka];
        acc0 = __builtin_amdgcn_wmma_f32_16x16x4_f32(false, a, false, b0, (short)0, acc0, false, false);
        acc1 = __builtin_amdgcn_wmma_f32_16x16x4_f32(false, a, false, b1, (short)0, acc1, false, false);
        acc2 = __builtin_amdgcn_wmma_f32_16x16x4_f32(false, a, false, b2, (short)0, acc2, false, false);
        acc3 = __builtin_amdgcn_wmma_f32_16x16x4_f32(false, a, false, b3, (short)0, acc3, false, false);
    }

    v8f accs[4] = { acc0, acc1, acc2, acc3 };
    float* Cw = C + (size_t)m0 * ldc + n0;
#pragma unroll
    for (int nt = 0; nt < 4; ++nt) {
        const int col = nt * 16 + r;
        const float bv = HASBIAS ? bias[n0 + col] : 0.0f;
#pragma unroll
        for (int v = 0; v < 8; ++v) {
            float val = accs[nt][v] + bv;
            float* p = &Cw[(size_t)(v + 8 * g) * ldc + col];
            if (ACCUM) *p += val;
            else       *p = val;
        }
    }
}

// ---------------- edge-attention precompute ----------------
// wea[k,h] = sum_c We[k, h*DIM+c] * a_edge[h,c]   (16x3)
__global__ void k_wea(const float* __restrict__ We, const float* __restrict__ a_edge,
                      float* __restrict__ wea)
{
    int id = threadIdx.x;
    if (id >= EDIM * HH) return;
    int k = id / HH, h = id % HH;
    float s = 0.f;
    for (int c = 0; c < DIM; ++c)
        s += We[k * (HH * DIM) + h * DIM + c] * a_edge[h * DIM + c];
    wea[k * HH + h] = s;
}

// e_att[e,h] = sum_k edge_attr[e,k] * wea[k,h]
__global__ void k_eatt(const float* __restrict__ ea, const float* __restrict__ wea,
                       float* __restrict__ eatt)
{
    int e = blockIdx.x * blockDim.x + threadIdx.x;
    if (e >= EE) return;
    float s0 = 0.f, s1 = 0.f, s2 = 0.f;
    for (int k = 0; k < EDIM; ++k) {
        float v = ea[e * EDIM + k];
        s0 += v * wea[k * HH + 0];
        s1 += v * wea[k * HH + 1];
        s2 += v * wea[k * HH + 2];
    }
    eatt[e * HH + 0] = s0;
    eatt[e * HH + 1] = s1;
    eatt[e * HH + 2] = s2;
}

// s_src[n,h] = <xw[n,h,:], a_src[h,:]> ; likewise s_dst
__global__ void k_scores(const float* __restrict__ xw, const float* __restrict__ a_src,
                         const float* __restrict__ a_dst, float* __restrict__ ssrc,
                         float* __restrict__ sdst)
{
    int id = blockIdx.x * blockDim.x + threadIdx.x;
    if (id >= NN * HH) return;
    int n = id / HH, h = id % HH;
    const float* xr = xw + (size_t)n * (HH * DIM) + h * DIM;
    float ss = 0.f, sd = 0.f;
    for (int c = 0; c < DIM; ++c) {
        float v = xr[c];
        ss += v * a_src[h * DIM + c];
        sd += v * a_dst[h * DIM + c];
    }
    ssrc[id] = ss;
    sdst[id] = sd;
}

__global__ void k_logits(const int* __restrict__ src, const int* __restrict__ dst,
                         const float* __restrict__ ssrc, const float* __restrict__ sdst,
                         const float* __restrict__ eatt, float* __restrict__ elog,
                         float* __restrict__ amax)
{
    int id = blockIdx.x * blockDim.x + threadIdx.x;
    if (id >= EE * HH) return;
    int e = id / HH, h = id % HH;
    int d = dst[e];
    float s = ssrc[src[e] * HH + h] + sdst[d * HH + h] + eatt[id];
    s = (s > 0.f) ? s : 0.2f * s;           // leaky_relu 0.2
    elog[id] = s;
    atomicMaxF(&amax[d * HH + h], s);
}

__global__ void k_exp(const int* __restrict__ dst, float* __restrict__ elog,
                      const float* __restrict__ amax, float* __restrict__ denom)
{
    int id = blockIdx.x * blockDim.x + threadIdx.x;
    if (id >= EE * HH) return;
    int e = id / HH, h = id % HH;
    int d = dst[e];
    float ex = expf(elog[id] - amax[d * HH + h]);
    elog[id] = ex;
    atomicAdd(&denom[d * HH + h], ex);
}

// m[dst,c] += sum_h att[e,h] * xw[src,h,c]   (head mean folded in later)
__global__ void k_agg(const int* __restrict__ src, const int* __restrict__ dst,
                      const float* __restrict__ elog, const float* __restrict__ denom,
                      const float* __restrict__ xw, float* __restrict__ m)
{
    long id = (long)blockIdx.x * blockDim.x + threadIdx.x;
    if (id >= (long)EE * DIM) return;
    int e = (int)(id >> 6);
    int c = (int)(id & 63);
    int s = src[e], d = dst[e];
    float acc = 0.f;
#pragma unroll
    for (int h = 0; h < HH; ++h) {
        float att = elog[e * HH + h] / (denom[d * HH + h] + 1e-16f);
        acc += att * xw[(size_t)s * (HH * DIM) + h * DIM + c];
    }
    atomicAdd(&m[d * DIM + c], acc);
}

// m = celu(m/H + bg)
__global__ void k_celu(float* __restrict__ m, const float* __restrict__ bg)
{
    int id = blockIdx.x * blockDim.x + threadIdx.x;
    if (id >= NN * DIM) return;
    float v = m[id] * (1.0f / HH) + bg[id & 63];
    m[id] = (v > 0.f) ? v : expm1f(v);
}

// GRU pointwise: gi/gh already include biases (added in GEMM)
__global__ void k_gru(const float* __restrict__ gi, const float* __restrict__ gh,
                      const float* __restrict__ h, float* __restrict__ hn)
{
    int id = blockIdx.x * blockDim.x + threadIdx.x;
    if (id >= NN * DIM) return;
    int n = id >> 6, c = id & 63;
    const float* gir = gi + (size_t)n * (3 * DIM);
    const float* ghr = gh + (size_t)n * (3 * DIM);
    float r  = 1.f / (1.f + expf(-(gir[c]           + ghr[c])));
    float z  = 1.f / (1.f + expf(-(gir[DIM + c]     + ghr[DIM + c])));
    float nn = tanhf(gir[2 * DIM + c] + r * ghr[2 * DIM + c]);
    hn[id] = (1.f - z) * nn + z * h[id];
}

__global__ void k_bnred(const float* __restrict__ h, float* __restrict__ stats)
{
    int id = blockIdx.x * blockDim.x + threadIdx.x;
    if (id >= NN * DIM) return;
    float v = h[id];
    atomicAdd(&stats[id & 63], v);
    atomicAdd(&stats[64 + (id & 63)], v * v);
}

__global__ void k_bnapply(const float* __restrict__ h, const float* __restrict__ stats,
                          const float* __restrict__ gamma, const float* __restrict__ beta,
                          float* __restrict__ x)
{
    int id = blockIdx.x * blockDim.x + threadIdx.x;
    if (id >= NN * DIM) return;
    int c = id & 63;
    float mu  = stats[c] * (1.f / NN);
    float var = stats[64 + c] * (1.f / NN) - mu * mu;
    x[id] = (h[id] - mu) * rsqrtf(var + 1e-5f) * gamma[c] + beta[c];
}

__global__ void k_outinit(float* __restrict__ out, const float* __restrict__ blin)
{
    int id = blockIdx.x * blockDim.x + threadIdx.x;
    if (id >= NN * DIM) return;
    out[id] = blin[id & 63];
}

// ---------------- launch ----------------
extern "C" void kernel_launch(void* const* d_in, const int* in_sizes, int n_in,
                              void* d_out, int out_size, void* d_ws, size_t ws_size,
                              hipStream_t stream)
{
    const float* x      = (const float*)d_in[0];
    const int*   ei     = (const int*)  d_in[1];
    const float* ea     = (const float*)d_in[2];
    const float* Wg     = (const float*)d_in[3];
    const float* bg     = (const float*)d_in[4];
    const float* a_src  = (const float*)d_in[5];
    const float* a_dst  = (const float*)d_in[6];
    const float* We     = (const float*)d_in[7];
    const float* a_edge = (const float*)d_in[8];
    const float* Wih    = (const float*)d_in[9];
    const float* Whh    = (const float*)d_in[10];
    const float* bih    = (const float*)d_in[11];
    const float* bhh    = (const float*)d_in[12];
    const float* gamma  = (const float*)d_in[13];
    const float* beta   = (const float*)d_in[14];
    const float* Wlin   = (const float*)d_in[15];
    const float* blin   = (const float*)d_in[16];
    const int* src = ei;
    const int* dst = ei + EE;

    float* ws = (float*)d_ws;
    float* xw    = ws; ws += (size_t)NN * HH * DIM;   // also reused as gi
    float* gh    = ws; ws += (size_t)NN * HH * DIM;
    float* ssrc  = ws; ws += NN * HH;
    float* sdst  = ws; ws += NN * HH;
    float* eatt  = ws; ws += (size_t)EE * HH;
    float* elog  = ws; ws += (size_t)EE * HH;
    float* amax  = ws; ws += NN * HH;
    float* denom = ws; ws += NN * HH;
    float* m     = ws; ws += NN * DIM;
    float* hA    = ws; ws += NN * DIM;
    float* hB    = ws; ws += NN * DIM;
    float* xA    = ws; ws += NN * DIM;
    float* xB    = ws; ws += NN * DIM;
    float* stats = ws; ws += 128;
    float* wea   = ws; ws += EDIM * HH;
    float* out = (float*)d_out;

    const dim3 gBlk(MWAVES * 32);                 // 5 waves
    const dim3 gGrid192(NN / 16 / MWAVES, 3);     // 64-col strips of a 192-wide C
    const dim3 gGrid64 (NN / 16 / MWAVES, 1);     // 64-wide C
    auto nb = [](long n) { return dim3((unsigned)((n + 255) / 256)); };

    // once-per-call precompute
    k_wea  <<<1, 64, 0, stream>>>(We, a_edge, wea);
    k_eatt <<<nb(EE), 256, 0, stream>>>(ea, wea, eatt);
    k_outinit<<<nb((long)NN * DIM), 256, 0, stream>>>(out, blin);
    // out += feats[0] @ Wlin[0:64,:]
    k_wmma_gemm<false, true, false><<<gGrid64, gBlk, 0, stream>>>(
        x, Wlin, nullptr, out, DIM, DIM);

    const float* xcur = x;
    const float* hcur = x;
    for (int t = 0; t < TT; ++t) {
        float* hnew = (t & 1) ? hB : hA;
        float* xnew = (t & 1) ? xB : xA;

        // xw = xcur @ Wg   [N,64]@[64,192]
        k_wmma_gemm<false, false, false><<<gGrid192, gBlk, 0, stream>>>(
            xcur, Wg, nullptr, xw, HH * DIM, HH * DIM);

        k_scores<<<nb((long)NN * HH), 256, 0, stream>>>(xw, a_src, a_dst, ssrc, sdst);
        k_fill<<<nb((long)NN * HH), 256, 0, stream>>>(amax, -INFINITY, NN * HH);
        k_fill<<<nb((long)NN * HH), 256, 0, stream>>>(denom, 0.f, NN * HH);
        k_fill<<<nb((long)NN * DIM), 256, 0, stream>>>(m, 0.f, NN * DIM);

        k_logits<<<nb((long)EE * HH), 256, 0, stream>>>(src, dst, ssrc, sdst, eatt, elog, amax);
        k_exp   <<<nb((long)EE * HH), 256, 0, stream>>>(dst, elog, amax, denom);
        k_agg   <<<nb((long)EE * DIM), 256, 0, stream>>>(src, dst, elog, denom, xw, m);
        k_celu  <<<nb((long)NN * DIM), 256, 0, stream>>>(m, bg);

        // gi = m @ Wih^T + bih (into xw buffer), gh = h @ Whh^T + bhh
        k_wmma_gemm<true, false, true><<<gGrid192, gBlk, 0, stream>>>(
            m, Wih, bih, xw, DIM, 3 * DIM);
        k_wmma_gemm<true, false, true><<<gGrid192, gBlk, 0, stream>>>(
            hcur, Whh, bhh, gh, DIM, 3 * DIM);
        k_gru<<<nb((long)NN * DIM), 256, 0, stream>>>(xw, gh, hcur, hnew);

        // node BatchNorm
        k_fill   <<<1, 128, 0, stream>>>(stats, 0.f, 128);
        k_bnred  <<<nb((long)NN * DIM), 256, 0, stream>>>(hnew, stats);
        k_bnapply<<<nb((long)NN * DIM), 256, 0, stream>>>(
            hnew, stats, gamma + t * DIM, beta + t * DIM, xnew);

        // out += feats[t+1] @ Wlin[(t+1)*64 : (t+2)*64, :]
        k_wmma_gemm<false, true, false><<<gGrid64, gBlk, 0, stream>>>(
            xnew, Wlin + (size_t)(t + 1) * DIM * DIM, nullptr, out, DIM, DIM);

        xcur = xnew;
        hcur = hnew;
    }
}